// Weighted_Cross_Entropy_Loss_111669149713
// MI455X (gfx1250) — compile-verified
//
#include <hip/hip_runtime.h>
#include <stdint.h>

// Weighted cross-entropy loss, pred[16,8,512,512] f32, target[16,512,512] i32.
// Pass 0: zero 8 histogram bins in ws.
// Pass 1: histogram of target (async-to-LDS staging, packed u64 counters, int atomics).
// Pass 2: per-pixel log-softmax + gather + weighted accumulate -> per-block partials.
// Pass 3: single-wave WMMA row-sum reduction of partials -> scalar loss.

#define HW_PIX 262144            // 512*512
#define NCLS 8
#define NIMG 16
#define BLOCKS_PER_IMG 128       // 262144 / 2048
#define PIX_PER_BLOCK 2048
#define THREADS 256

typedef __attribute__((ext_vector_type(2))) float v2f;
typedef __attribute__((ext_vector_type(8))) float v8f;

// Pointee type the async-LDS builtin expects: int __vector_size__(16)
typedef int v4i_t __attribute__((vector_size(16)));
typedef __attribute__((address_space(1))) v4i_t gv4i_t;   // global
typedef __attribute__((address_space(3))) v4i_t lv4i_t;   // LDS

__device__ __forceinline__ gv4i_t* to_g4(const void* p) {
  return (gv4i_t*)(uintptr_t)p;
}
__device__ __forceinline__ lv4i_t* to_l4(const void* p) {
  // generic LDS pointer: low 32 bits are the LDS byte offset (aperture layout)
  return (lv4i_t*)(uint32_t)(uintptr_t)p;
}

__global__ void wce_zero(unsigned* __restrict__ counts) {
  if (threadIdx.x < NCLS) counts[threadIdx.x] = 0u;
}

// ---------------------------------------------------------------------------
// Pass 1: histogram. 512 blocks x 256 threads x 32 ints = 4,194,304 elements.
// Target tile (32 KB/block) staged to LDS with GLOBAL_LOAD_ASYNC_TO_LDS_B128.
// Per-thread counts packed as 4x16-bit fields in two u64s (exact, deterministic).
// ---------------------------------------------------------------------------
__global__ __launch_bounds__(THREADS) void wce_hist(const int* __restrict__ target,
                                                    unsigned* __restrict__ counts) {
  __shared__ __align__(16) int tile[THREADS * 32];   // 32 KB
  const int t = threadIdx.x;
  const int* gbase = target + (size_t)blockIdx.x * (THREADS * 32);

#if __has_builtin(__builtin_amdgcn_global_load_async_to_lds_b128)
  #pragma unroll
  for (int i = 0; i < 8; ++i) {
    const int off = i * 1024 + t * 4;  // int index; 16B per lane, 4KB contiguous per block-iter
    __builtin_amdgcn_global_load_async_to_lds_b128(to_g4(gbase + off),
                                                   to_l4(&tile[off]), 0, 0);
  }
  #if __has_builtin(__builtin_amdgcn_s_wait_asynccnt)
  __builtin_amdgcn_s_wait_asynccnt(0);
  #else
  asm volatile("s_wait_asynccnt 0" ::: "memory");
  #endif
#else
  #pragma unroll
  for (int i = 0; i < 8; ++i) {
    const int off = i * 1024 + t * 4;
    *(int4*)&tile[off] = *(const int4*)(gbase + off);
  }
#endif
  __syncthreads();

  unsigned long long acc0 = 0ull, acc1 = 0ull;  // bins 0-3 / 4-7, 16-bit fields
  #pragma unroll
  for (int k = 0; k < 8; ++k) {
    int4 v4 = *(const int4*)&tile[t * 32 + k * 4];
    int vv[4] = {v4.x, v4.y, v4.z, v4.w};
    #pragma unroll
    for (int j = 0; j < 4; ++j) {
      int v = vv[j];
      v = v < 0 ? 0 : (v > 7 ? 7 : v);
      unsigned long long one = 1ull << ((v & 3) * 16);
      acc0 += (v < 4) ? one : 0ull;
      acc1 += (v < 4) ? 0ull : one;
    }
  }

  // Block max per bin = 256*32 = 8192 < 65536: 16-bit fields cannot overflow.
  __shared__ unsigned long long red[THREADS];
  red[t] = acc0;
  __syncthreads();
  for (int s = THREADS / 2; s > 0; s >>= 1) {
    if (t < s) red[t] += red[t + s];
    __syncthreads();
  }
  unsigned long long tot0 = red[0];
  __syncthreads();
  red[t] = acc1;
  __syncthreads();
  for (int s = THREADS / 2; s > 0; s >>= 1) {
    if (t < s) red[t] += red[t + s];
    __syncthreads();
  }
  unsigned long long tot1 = red[0];

  if (t == 0) {
    #pragma unroll
    for (int i = 0; i < 4; ++i) {
      atomicAdd(&counts[i],     (unsigned)((tot0 >> (i * 16)) & 0xFFFFull));
      atomicAdd(&counts[4 + i], (unsigned)((tot1 >> (i * 16)) & 0xFFFFull));
    }
  }
}

// ---------------------------------------------------------------------------
// Pass 2: 2048 blocks (16 images x 128); each thread handles 8 consecutive
// pixels: 16x float4 pred loads + 2x int4 target loads, fully unrolled c=8
// softmax (register-only; target-class gather via compare-select).
// ---------------------------------------------------------------------------
__global__ __launch_bounds__(THREADS) void wce_main(const float* __restrict__ pred,
                                                    const int* __restrict__ target,
                                                    const unsigned* __restrict__ counts,
                                                    float* __restrict__ partials) {
  const int t = threadIdx.x;
  const int n = blockIdx.x >> 7;
  const int b = blockIdx.x & 127;

  __shared__ float s_invc[NCLS];
  if (t < NCLS) s_invc[t] = 1.0f / ((float)counts[t] + 1e-6f);
  __syncthreads();

  const size_t pixBase = (size_t)b * PIX_PER_BLOCK + (size_t)t * 8;
  const float* p = pred + (size_t)n * NCLS * HW_PIX + pixBase;
  const int* tg = target + (size_t)n * HW_PIX + pixBase;

  int4 ta = *(const int4*)tg;
  int4 tb2 = *(const int4*)(tg + 4);
  int tgt[8] = {ta.x, ta.y, ta.z, ta.w, tb2.x, tb2.y, tb2.z, tb2.w};

  float xr[NCLS][8];
  #pragma unroll
  for (int c = 0; c < NCLS; ++c) {
    float4 a = *(const float4*)(p + (size_t)c * HW_PIX);
    float4 d = *(const float4*)(p + (size_t)c * HW_PIX + 4);
    xr[c][0] = a.x; xr[c][1] = a.y; xr[c][2] = a.z; xr[c][3] = a.w;
    xr[c][4] = d.x; xr[c][5] = d.y; xr[c][6] = d.z; xr[c][7] = d.w;
  }

  float num = 0.0f, den = 0.0f;
  #pragma unroll
  for (int j = 0; j < 8; ++j) {
    int tv = tgt[j];
    tv = tv < 0 ? 0 : (tv > 7 ? 7 : tv);
    float m = xr[0][j];
    #pragma unroll
    for (int c = 1; c < NCLS; ++c) m = fmaxf(m, xr[c][j]);
    float s = 0.0f;
    float xt = xr[0][j];
    #pragma unroll
    for (int c = 0; c < NCLS; ++c) {
      s += __expf(xr[c][j] - m);
      xt = (c == tv) ? xr[c][j] : xt;   // cndmask gather, no scratch
    }
    float lp = xt - m - __logf(s);
    float w = s_invc[tv];
    num = fmaf(lp, w, num);
    den += w;
  }

  __shared__ float rn[THREADS], rd[THREADS];
  rn[t] = num;
  rd[t] = den;
  __syncthreads();
  for (int s2 = THREADS / 2; s2 > 0; s2 >>= 1) {
    if (t < s2) { rn[t] += rn[t + s2]; rd[t] += rd[t + s2]; }
    __syncthreads();
  }
  if (t == 0) {
    partials[n * BLOCKS_PER_IMG + b] = rn[0];                 // num partial
    partials[(NIMG + n) * BLOCKS_PER_IMG + b] = rd[0];        // den partial
  }
}

// ---------------------------------------------------------------------------
// Pass 3: one wave32. Row-sums of partials[2][16][128] via
// V_WMMA_F32_16X16X4_F32: D += A(16x4 chunk) x B(4x16 all-ones).
// A layout (ISA 7.12.2): lanes 0-15 hold K=0,1; lanes 16-31 hold K=2,3; M=lane%16.
// D layout: lane 0 vgpr r = D[r][0]; lane 16 vgpr r = D[8+r][0].
// ---------------------------------------------------------------------------
__global__ __launch_bounds__(32) void wce_final(const float* __restrict__ partials,
                                                float* __restrict__ out) {
  const int lane = threadIdx.x;
  const int row = lane & 15;
  const int koff = (lane >> 4) * 2;

  __shared__ float s_num[NIMG], s_den[NIMG];

#if __has_builtin(__builtin_amdgcn_wmma_f32_16x16x4_f32)
  v8f dn = {0.f, 0.f, 0.f, 0.f, 0.f, 0.f, 0.f, 0.f};
  v8f dd = {0.f, 0.f, 0.f, 0.f, 0.f, 0.f, 0.f, 0.f};
  v2f bones; bones.x = 1.0f; bones.y = 1.0f;   // B = ones: layout-independent
  for (int kb = 0; kb < BLOCKS_PER_IMG; kb += 4) {
    v2f an, ad;
    an.x = partials[row * BLOCKS_PER_IMG + kb + koff];
    an.y = partials[row * BLOCKS_PER_IMG + kb + koff + 1];
    ad.x = partials[(NIMG + row) * BLOCKS_PER_IMG + kb + koff];
    ad.y = partials[(NIMG + row) * BLOCKS_PER_IMG + kb + koff + 1];
    dn = __builtin_amdgcn_wmma_f32_16x16x4_f32(false, an, false, bones,
                                               (short)0, dn, false, false);
    dd = __builtin_amdgcn_wmma_f32_16x16x4_f32(false, ad, false, bones,
                                               (short)0, dd, false, false);
  }
  if ((lane & 15) == 0) {
    const int base = (lane >> 4) * 8;
    #pragma unroll
    for (int r = 0; r < 8; ++r) {
      s_num[base + r] = dn[r];
      s_den[base + r] = dd[r];
    }
  }
#else
  if (lane < NIMG) {
    float sn = 0.f, sd = 0.f;
    for (int k = 0; k < BLOCKS_PER_IMG; ++k) {
      sn += partials[lane * BLOCKS_PER_IMG + k];
      sd += partials[(NIMG + lane) * BLOCKS_PER_IMG + k];
    }
    s_num[lane] = sn;
    s_den[lane] = sd;
  }
#endif
  __syncthreads();
  if (lane == 0) {
    float acc = 0.f;
    #pragma unroll
    for (int i = 0; i < NIMG; ++i) acc += s_num[i] / s_den[i];
    out[0] = -acc * (1.0f / (float)NIMG);
  }
}

extern "C" void kernel_launch(void* const* d_in, const int* in_sizes, int n_in,
                              void* d_out, int out_size, void* d_ws, size_t ws_size,
                              hipStream_t stream) {
  const float* pred = (const float*)d_in[0];
  const int* target = (const int*)d_in[1];
  unsigned* counts = (unsigned*)d_ws;                       // 8 x u32
  float* partials = (float*)((char*)d_ws + 64);             // [2][16][128] f32
  float* out = (float*)d_out;

  wce_zero<<<1, 32, 0, stream>>>(counts);
  wce_hist<<<512, THREADS, 0, stream>>>(target, counts);
  wce_main<<<NIMG * BLOCKS_PER_IMG, THREADS, 0, stream>>>(pred, target, counts, partials);
  wce_final<<<1, 32, 0, stream>>>(partials, out);
}